// extend_MipRayMarcher_21225728376848
// MI455X (gfx1250) — compile-verified
//
#include <hip/hip_runtime.h>
#include <cstdint>
#include <cstddef>

// Problem constants (from reference): B=4, R=4096, S=48, C=32
#define NB    4
#define NRAYS 4096
#define NS    48
#define NC    32
#define NMID  (NS - 1)          // 47
#define NRTOT (NB * NRAYS)      // 16384 rays
#define WAVES 8                 // waves per block (wave32) -> 256 threads

// stable softplus via hardware transcendentals:
// log1p(exp(x)) = max(x,0) + log(1 + exp(-|x|))
__device__ __forceinline__ float softplus_f(float x) {
  return fmaxf(x, 0.0f) + __logf(1.0f + __expf(-fabsf(x)));
}

// CDNA5 async copy: 16B per lane, global -> LDS, tracked by ASYNCcnt.
// The same literal offset is applied to both the LDS and global addresses
// (CDNA5 ISA), so one addr pair serves the whole tile.
#define ASYNC_CP16(ldsaddr, gaddr, OFF)                                      \
  asm volatile("global_load_async_to_lds_b128 %0, %1, off offset:" #OFF      \
               ::"v"(ldsaddr), "v"(gaddr) : "memory")

__global__ __launch_bounds__(WAVES * 32)
void raymarch_composite_kernel(const float* __restrict__ colors,     // [B,R,S,C]
                               const float* __restrict__ densities,  // [B,R,S,1]
                               const float* __restrict__ depths,     // [B,R,S,1]
                               float* __restrict__ out_rgb,          // [B,R,C]
                               float* __restrict__ out_depth,        // [B,R,1]
                               float* __restrict__ out_w,            // [B,R,S-1,1]
                               float* __restrict__ out_wtot) {       // [B,R,1]
  __shared__ float s_col[WAVES][NS * NC];  // 6144 B per wave color tile
  __shared__ float s_dep[WAVES][NS];
  __shared__ float s_den[WAVES][NS];

  const int lane = threadIdx.x & 31;
  const int w    = threadIdx.x >> 5;
  const int ray  = blockIdx.x * WAVES + w;   // grid sized exactly: 0..16383

  // ---- 1) kick off async DMA of this ray's 48x32 f32 color tile to LDS ----
  // 48 rows * 128 B = 6144 B; each instruction moves 32 lanes * 16 B = 512 B.
  {
    const char* g = (const char*)(colors + (size_t)ray * (NS * NC)) + lane * 16;
    uint32_t l = (uint32_t)(uintptr_t)(&s_col[w][0]) + (uint32_t)(lane * 16);
    ASYNC_CP16(l, g, 0);
    ASYNC_CP16(l, g, 512);
    ASYNC_CP16(l, g, 1024);
    ASYNC_CP16(l, g, 1536);
    ASYNC_CP16(l, g, 2048);
    ASYNC_CP16(l, g, 2560);
    ASYNC_CP16(l, g, 3072);
    ASYNC_CP16(l, g, 3584);
    ASYNC_CP16(l, g, 4096);
    ASYNC_CP16(l, g, 4608);
    ASYNC_CP16(l, g, 5120);
    ASYNC_CP16(l, g, 5632);
  }

  // ---- 2) stage per-ray scalars (48 depths + 48 densities), streaming ----
  const size_t sbase = (size_t)ray * NS;
  s_dep[w][lane] = __builtin_nontemporal_load(depths + sbase + lane);
  s_den[w][lane] = __builtin_nontemporal_load(densities + sbase + lane);
  if (lane < NS - 32) {  // lanes 0..15 cover samples 32..47
    s_dep[w][32 + lane] = __builtin_nontemporal_load(depths + sbase + 32 + lane);
    s_den[w][32 + lane] = __builtin_nontemporal_load(densities + sbase + 32 + lane);
  }
  __builtin_amdgcn_wave_barrier();
  asm volatile("s_wait_dscnt 0" ::: "memory");     // scalars visible wave-wide
  asm volatile("s_wait_asynccnt 0" ::: "memory");  // color tile resident in LDS

  // ---- 3) serial transmittance chain + per-lane channel accumulation ----
  float T    = 1.0f;   // exclusive transmittance (cumprod)
  float acc  = 0.0f;   // weighted color, channel = lane
  float accd = 0.0f;   // weighted depth
  float accw = 0.0f;   // weight total
  float w_a  = 0.0f;   // captured weight for sample index == lane
  float w_b  = 0.0f;   // captured weight for sample index == lane + 32
  float cprev = s_col[w][lane];
  float d0 = s_dep[w][0];
  float q0 = s_den[w][0];
  for (int s = 0; s < NMID; ++s) {
    const float d1    = s_dep[w][s + 1];
    const float q1    = s_den[w][s + 1];
    const float delta = d1 - d0;
    const float dmid  = 0.5f * (q0 + q1) - 1.0f;        // densities_mid - 1
    const float sig   = softplus_f(dmid);               // clamp_mode softplus
    const float alpha = 1.0f - __expf(-sig * delta);
    const float wt    = alpha * T;
    T *= (1.0f - alpha) + 1e-10f;

    // every lane computes the same wt -> capture cross-lane in registers so
    // the final weight stores are coalesced with no LDS round-trip
    if (s == lane)      w_a = wt;
    if (s == lane + 32) w_b = wt;   // only lanes 0..14 ever match

    const float cnext = s_col[w][(s + 1) * NC + lane];
    acc  = fmaf(wt, 0.5f * (cprev + cnext), acc);
    accd = fmaf(wt, 0.5f * (d0 + d1), accd);
    accw += wt;
    cprev = cnext; d0 = d1; q0 = q1;
  }

  // ---- 4) streaming stores (nothing is ever re-read: bypass-rinse L2) ----
  __builtin_nontemporal_store(acc * 2.0f - 1.0f, out_rgb + (size_t)ray * NC + lane);
  if (lane == 0) {
    __builtin_nontemporal_store(accd / (accw + 0.001f), out_depth + ray);
    __builtin_nontemporal_store(accw, out_wtot + ray);
  }
  const size_t wb = (size_t)ray * NMID;
  __builtin_nontemporal_store(w_a, out_w + wb + lane);        // weights 0..31
  if (lane < NMID - 32)                                       // lanes 0..14
    __builtin_nontemporal_store(w_b, out_w + wb + 32 + lane); // weights 32..46
}

extern "C" void kernel_launch(void* const* d_in, const int* in_sizes, int n_in,
                              void* d_out, int out_size, void* d_ws, size_t ws_size,
                              hipStream_t stream) {
  (void)in_sizes; (void)n_in; (void)out_size; (void)d_ws; (void)ws_size;

  const float* colors    = (const float*)d_in[0];  // [4,4096,48,32] f32
  const float* densities = (const float*)d_in[1];  // [4,4096,48,1]  f32
  const float* depths    = (const float*)d_in[2];  // [4,4096,48,1]  f32

  // Outputs concatenated flat in reference return order.
  float* out       = (float*)d_out;
  float* out_rgb   = out;                                   // 16384*32
  float* out_depth = out_rgb + (size_t)NRTOT * NC;          // 16384
  float* out_w     = out_depth + NRTOT;                     // 16384*47
  float* out_wtot  = out_w + (size_t)NRTOT * NMID;          // 16384

  const int threads = WAVES * 32;                 // 256
  const int blocks  = NRTOT / WAVES;              // 2048 (exact)
  raymarch_composite_kernel<<<blocks, threads, 0, stream>>>(
      colors, densities, depths, out_rgb, out_depth, out_w, out_wtot);
}